// BMMRemapper_73959336837140
// MI455X (gfx1250) — compile-verified
//
#include <hip/hip_runtime.h>

// Bilinear grid-resample (the reference's sparse "interp matrix" bmm collapsed
// to its 4-nnz-per-row gather form). CDNA5 path: async global->LDS staging of
// x channel planes, LDS gathers for the 4 corners, nontemporal output stores.

namespace {
constexpr int Nn      = 8;
constexpr int Hh      = 48;
constexpr int Ww      = 48;
constexpr int HW      = Hh * Ww;      // 2304
constexpr int Cc      = 128;
constexpr int CB      = 4;            // channels staged per block (36 KB LDS)
constexpr int CBLKS   = Cc / CB;      // 32
constexpr int NBLK    = Nn * CBLKS;   // 256 workgroups
constexpr int THREADS = 256;          // 8 waves (wave32)
constexpr float EPSV  = 1e-5f;
} // namespace

typedef __attribute__((ext_vector_type(4))) int v4i;

#if defined(__HIP_DEVICE_COMPILE__)
#if __has_builtin(__builtin_amdgcn_global_load_async_to_lds_b128)
#define HAVE_ASYNC_LDS 1
#endif
#if __has_builtin(__builtin_amdgcn_s_wait_asynccnt)
#define HAVE_WAIT_ASYNC 1
#endif
#endif

__global__ __launch_bounds__(THREADS) void bilinear_gather_kernel(
    const float* __restrict__ x,     // (N, C, H, W)
    const float* __restrict__ grid,  // (N, H, W, 2)
    float* __restrict__ out)         // (N, C, H, W)
{
  extern __shared__ float lds[];     // CB * HW floats

  const int blk = blockIdx.x;
  const int n   = blk / CBLKS;
  const int c0  = (blk % CBLKS) * CB;
  const int tid = threadIdx.x;

  // ---- Stage x[n, c0:c0+CB, :] into LDS (async global->LDS copies) ----
  const float4* __restrict__ src =
      (const float4*)(x + ((size_t)n * Cc + c0) * HW);
  float4* dst4 = (float4*)lds;
  constexpr int NV4 = CB * HW / 4;   // 2304 16B packets, 9 per thread

#if HAVE_ASYNC_LDS
  for (int i = tid; i < NV4; i += THREADS) {
    __builtin_amdgcn_global_load_async_to_lds_b128(
        (__attribute__((address_space(1))) v4i*)(src + i),
        (__attribute__((address_space(3))) v4i*)(dst4 + i),
        /*imm offset=*/0, /*cpol=*/0);
  }
#if HAVE_WAIT_ASYNC
  __builtin_amdgcn_s_wait_asynccnt(0);
#else
  asm volatile("s_wait_asynccnt 0x0" ::: "memory");
#endif
#else
  for (int i = tid; i < NV4; i += THREADS) dst4[i] = src[i];
#endif
  __syncthreads();

  // ---- Per-output-pixel gather ----
  const float2* __restrict__ g2 = (const float2*)grid;  // (N, HW) float2

  for (int q = tid; q < HW; q += THREADS) {
    // Disk mask: valid iff inside [-0.5, dim-0.5] for ALL batch elements.
    bool ok = true;
#pragma unroll
    for (int nn = 0; nn < Nn; ++nn) {
      float2 g = g2[nn * HW + q];
      ok = ok && (g.x >= -0.5f) && (g.x <= (float)Hh - 0.5f) &&
           (g.y >= -0.5f) && (g.y <= (float)Ww - 0.5f);
    }

    float2 g  = g2[n * HW + q];
    float gy  = fminf(fmaxf(g.x, EPSV), (float)(Hh - 1) - EPSV);
    float gx  = fminf(fmaxf(g.y, EPSV), (float)(Ww - 1) - EPSV);
    float by  = floorf(gy);
    float bx  = floorf(gx);
    int   lin = (int)by * Ww + (int)bx;   // clamped => lin+49 <= HW-1
    float fa  = gy - by;
    float fb  = gx - bx;
    float m   = ok ? 1.0f : 0.0f;
    float w00 = m * (1.0f - fa) * (1.0f - fb);
    float w01 = m * (1.0f - fa) * fb;
    float w10 = m * fa * (1.0f - fb);
    float w11 = m * fa * fb;

    const size_t obase = ((size_t)n * Cc + c0) * HW + (size_t)q;
#pragma unroll
    for (int cc = 0; cc < CB; ++cc) {
      const float* p = lds + cc * HW;
      float v = w00 * p[lin] + w01 * p[lin + 1] +
                w10 * p[lin + Ww] + w11 * p[lin + Ww + 1];
      __builtin_nontemporal_store(v, out + obase + (size_t)cc * HW);
    }
  }
}

extern "C" void kernel_launch(void* const* d_in, const int* in_sizes, int n_in,
                              void* d_out, int out_size, void* d_ws,
                              size_t ws_size, hipStream_t stream) {
  const float* x    = (const float*)d_in[0];  // (8,128,48,48) f32
  const float* grid = (const float*)d_in[1];  // (8,48,48,2)   f32
  float*       out  = (float*)d_out;          // (8,128,48,48) f32
  (void)in_sizes; (void)n_in; (void)out_size; (void)d_ws; (void)ws_size;

  const size_t shmem = (size_t)CB * HW * sizeof(float);  // 36 KB
  hipLaunchKernelGGL(bilinear_gather_kernel, dim3(NBLK), dim3(THREADS), shmem,
                     stream, x, grid, out);
}